// BodySegment_47356309405697
// MI455X (gfx1250) — compile-verified
//
#include <hip/hip_runtime.h>
#include <math.h>

typedef __attribute__((ext_vector_type(2))) float v2f;
typedef __attribute__((ext_vector_type(8))) float v8f;

#define NB 4
#define NV 6890
#define NP 1024
#define NF 2048
#define LBAND 64
#define NTILE (NF / 16)   /* 128 triangle tiles */
#define PTILES (NP / 16)  /* 64 point tiles per batch */
#define TBL_OFF 64        /* float offset of triangle table in ws */
#define TILE_FLOATS 448   /* 7 quantities * 64 floats = 1792 B per tile */

// ---------------------------------------------------------------------------
// Kernel 1: band means  b0[b][3], b1[b][3]  -> ws[band*12 + b*3 + c]
// ---------------------------------------------------------------------------
__global__ __launch_bounds__(32)
void band_means_kernel(const float* __restrict__ vtx,
                       const int* __restrict__ b0i,
                       const int* __restrict__ b1i,
                       float* __restrict__ ws)
{
    int t = threadIdx.x;
    if (t >= 24) return;
    int band = t / 12;
    int rem  = t % 12;
    int b    = rem / 3;
    int c    = rem % 3;
    const int* idx = band ? b1i : b0i;
    float s = 0.0f;
    for (int j = 0; j < LBAND; ++j) {
        int vi = idx[j];
        s += vtx[(b * NV + vi) * 3 + c];
    }
    ws[band * 12 + b * 3 + c] = s * (1.0f / LBAND);
}

// ---------------------------------------------------------------------------
// Kernel 2: per-triangle affine coefficient table (see round-0 derivation).
// Stored pre-swizzled for the WMMA B operand:
//    tbl[((b*NTILE + tile)*7 + q)*64 + half*32 + n*2 + {0,1}]
// ---------------------------------------------------------------------------
__device__ __forceinline__ float dot3(const float* a, const float* b) {
    return a[0]*b[0] + a[1]*b[1] + a[2]*b[2];
}
__device__ __forceinline__ void cross3(const float* a, const float* b, float* o) {
    o[0] = a[1]*b[2] - a[2]*b[1];
    o[1] = a[2]*b[0] - a[0]*b[2];
    o[2] = a[0]*b[1] - a[1]*b[0];
}

__global__ __launch_bounds__(256)
void tri_table_kernel(const float* __restrict__ vtx,
                      const int* __restrict__ faces,
                      float* __restrict__ ws)
{
    int gid = blockIdx.x * blockDim.x + threadIdx.x;
    if (gid >= NB * NF) return;
    int b = gid / NF;
    int f = gid % NF;

    float P[3][3];
#pragma unroll
    for (int k = 0; k < 3; ++k) {
        int vi = faces[f * 3 + k];
        if (vi < NV) {
#pragma unroll
            for (int c = 0; c < 3; ++c) P[k][c] = vtx[(b * NV + vi) * 3 + c];
        } else {
            int band = vi - NV; // 0 or 1
#pragma unroll
            for (int c = 0; c < 3; ++c) P[k][c] = ws[band * 12 + b * 3 + c];
        }
    }
    const float* A = P[0];
    const float* B = P[1];
    const float* C = P[2];

    float ab[3], bc[3], ca[3], n3[3];
    cross3(A, B, ab);
    cross3(B, C, bc);
    cross3(C, A, ca);
#pragma unroll
    for (int c = 0; c < 3; ++c) n3[c] = ab[c] + bc[c] + ca[c];
    float d = dot3(A, bc); // det(A,B,C)

    float* tbl = ws + TBL_OFF;
    int tile = f >> 4;
    int n2   = (f & 15) * 2;
    float* base = tbl + (size_t)(b * NTILE + tile) * TILE_FLOATS;

#define EMIT(q, t0, t1, t2, t3)                    \
    {                                              \
        float* p = base + (q) * 64;                \
        p[n2]        = (t0);                       \
        p[n2 + 1]    = (t1);                       \
        p[32 + n2]   = (t2);                       \
        p[32 + n2+1] = (t3);                       \
    }

    EMIT(0, -2.0f*A[0], -2.0f*A[1], -2.0f*A[2], dot3(A, A));
    EMIT(1, -2.0f*B[0], -2.0f*B[1], -2.0f*B[2], dot3(B, B));
    EMIT(2, -2.0f*C[0], -2.0f*C[1], -2.0f*C[2], dot3(C, C));
    EMIT(3, -(A[0]+B[0]), -(A[1]+B[1]), -(A[2]+B[2]), dot3(A, B));
    EMIT(4, -(B[0]+C[0]), -(B[1]+C[1]), -(B[2]+C[2]), dot3(B, C));
    EMIT(5, -(C[0]+A[0]), -(C[1]+A[1]), -(C[2]+A[2]), dot3(C, A));
    EMIT(6, -n3[0], -n3[1], -n3[2], d);
#undef EMIT
}

// ---------------------------------------------------------------------------
// Fast quadrant-correct atan2 (hardware rcp + 9th-order minimax, ~1e-5 rad).
// Matches jnp.arctan2 conventions: atan2(0,+x)=0, atan2(+-0,-x)=+-pi,
// atan2(0,0)=0.
// ---------------------------------------------------------------------------
__device__ __forceinline__ float fast_atan2f(float y, float x)
{
    float ay = fabsf(y), ax = fabsf(x);
    float hi = fmaxf(ay, ax);
    float lo = fminf(ay, ax);
    float t  = lo * __builtin_amdgcn_rcpf(hi);   // in [0,1]
    t = (hi == 0.0f) ? 0.0f : t;                 // atan2(0,0) -> 0
    float s = t * t;
    float r = fmaf(s, fmaf(s, fmaf(s, fmaf(s, 0.0208351f, -0.085133f),
                                   0.180141f), -0.3302995f), 0.9998660f);
    r = t * r;
    r = (ay > ax)   ? (1.5707963267948966f - r) : r;
    r = (x < 0.0f)  ? (3.1415926535897931f - r) : r;
    return copysignf(r, y);
}

// ---------------------------------------------------------------------------
// Kernel 3: winding numbers via V_WMMA_F32_16X16X4_F32 tiles.
// One wave per (batch, 16-point tile), 8 waves/block; all waves in a block
// share the same batch, so each 1792 B triangle tile is staged into LDS once
// per block with async global->LDS copies (double-buffered), then consumed
// by all 8 waves.
// ---------------------------------------------------------------------------
__global__ __launch_bounds__(256)
void wn_wmma_kernel(const float* __restrict__ vtx,
                    const int* __restrict__ svidx,
                    const float* __restrict__ ws,
                    float* __restrict__ out)
{
    __shared__ __align__(16) float lbuf[2 * TILE_FLOATS];

    const float* tbl = ws + TBL_OFF;
    int tid   = threadIdx.x;
    int lane  = tid & 31;
    int wave  = tid >> 5;
    int g     = blockIdx.x * 8 + wave;   // 0 .. NB*PTILES-1
    int b     = g / PTILES;              // uniform across the block
    int ptile = g % PTILES;
    int half  = lane >> 4;
    int mn    = lane & 15;

    // Point for row m = mn of this tile.
    int pidx = svidx[ptile * 16 + mn];
    float px = vtx[(b * NV + pidx) * 3 + 0];
    float py = vtx[(b * NV + pidx) * 3 + 1];
    float pz = vtx[(b * NV + pidx) * 3 + 2];
    float pp = px * px + py * py + pz * pz;

    v2f a1, a2, b2;
    a1.x = half ? pz : px;
    a1.y = half ? 1.0f : py;
    a2.x = half ? 0.0f : pp;     // second-K-group A: row coeff |p|^2
    a2.y = 0.0f;
    b2.x = half ? 0.0f : 1.0f;   // second-K-group B: ppcoef = 1
    b2.y = 0.0f;

    v8f zero = {};
    // pp_tile[m][n] = |p_m|^2 for every column; C-init for the 6 dot terms.
    v8f ppt = __builtin_amdgcn_wmma_f32_16x16x4_f32(
        false, a2, false, b2, (short)0, zero, false, false);

    // ---- async staging: 112 lanes copy 16 B each (1792 B per tile) -------
    const float* gbase = tbl + (size_t)b * NTILE * TILE_FLOATS;

    if (tid < 112) {  // prefetch tile 0 into buffer 0
        unsigned loff = (unsigned)(size_t)(const void*)&lbuf[tid * 4];
        unsigned long long ga =
            (unsigned long long)(size_t)(gbase + tid * 4);
        asm volatile("global_load_async_to_lds_b128 %0, %1, off"
                     :: "v"(loff), "v"(ga) : "memory");
    }
    asm volatile("s_wait_asynccnt 0x0" ::: "memory");
    __syncthreads();

    float acc[8] = {0.f, 0.f, 0.f, 0.f, 0.f, 0.f, 0.f, 0.f};

    for (int tile = 0; tile < NTILE; ++tile) {
        int cur = tile & 1;

        // issue prefetch of tile+1 into the other buffer
        if ((tile + 1 < NTILE) && (tid < 112)) {
            unsigned loff = (unsigned)(size_t)(const void*)
                                &lbuf[(cur ^ 1) * TILE_FLOATS + tid * 4];
            unsigned long long ga = (unsigned long long)(size_t)
                (gbase + (size_t)(tile + 1) * TILE_FLOATS + tid * 4);
            asm volatile("global_load_async_to_lds_b128 %0, %1, off"
                         :: "v"(loff), "v"(ga) : "memory");
        }

        int lbase = cur * TILE_FLOATS + half * 32 + mn * 2;
        v8f c[7];
#pragma unroll
        for (int q = 0; q < 7; ++q) {
            v2f bq;
            bq.x = lbuf[lbase + q * 64];
            bq.y = lbuf[lbase + q * 64 + 1];
            v8f ci = (q < 6) ? ppt : zero;  // det has no |p|^2 term
            c[q] = __builtin_amdgcn_wmma_f32_16x16x4_f32(
                false, a1, false, bq, (short)0, ci, false, false);
        }
#pragma unroll
        for (int v = 0; v < 8; ++v) {
            float la = __builtin_amdgcn_sqrtf(fmaxf(c[0][v], 0.0f));
            float lb = __builtin_amdgcn_sqrtf(fmaxf(c[1][v], 0.0f));
            float lc = __builtin_amdgcn_sqrtf(fmaxf(c[2][v], 0.0f));
            float denom = la * lb * lc + c[3][v] * lc
                        + c[4][v] * la + c[5][v] * lb;
            acc[v] += fast_atan2f(c[6][v], denom);
        }

        asm volatile("s_wait_asynccnt 0x0" ::: "memory");
        __syncthreads();
    }

    // Reduce over the 16 columns held by each lane half.
#pragma unroll
    for (int v = 0; v < 8; ++v) {
        float s = acc[v];
        s += __shfl_xor(s, 1, 32);
        s += __shfl_xor(s, 2, 32);
        s += __shfl_xor(s, 4, 32);
        s += __shfl_xor(s, 8, 32);
        acc[v] = s;
    }
    if (mn == 0) {
        // lane 0 writes rows 0..7, lane 16 writes rows 8..15
#pragma unroll
        for (int v = 0; v < 8; ++v) {
            // sum(2*atan2)/4pi == sum(atan2) * 1/(2pi)
            out[b * NP + ptile * 16 + half * 8 + v] =
                acc[v] * 0.15915494309189535f;
        }
    }
}

// ---------------------------------------------------------------------------
extern "C" void kernel_launch(void* const* d_in, const int* in_sizes, int n_in,
                              void* d_out, int out_size, void* d_ws, size_t ws_size,
                              hipStream_t stream)
{
    const float* vtx  = (const float*)d_in[0];  // [4,6890,3]
    const int* svidx  = (const int*)d_in[1];    // [1024]
    const int* b0i    = (const int*)d_in[2];    // [64]
    const int* b1i    = (const int*)d_in[3];    // [64]
    const int* faces  = (const int*)d_in[4];    // [2048,3]
    float* out = (float*)d_out;                 // [4,1024]
    float* ws  = (float*)d_ws;

    band_means_kernel<<<1, 32, 0, stream>>>(vtx, b0i, b1i, ws);
    tri_table_kernel<<<(NB * NF + 255) / 256, 256, 0, stream>>>(vtx, faces, ws);
    wn_wmma_kernel<<<(NB * PTILES) / 8, 256, 0, stream>>>(vtx, svidx, ws, out);
}